// GNN_76476187672878
// MI455X (gfx1250) — compile-verified
//
#include <hip/hip_runtime.h>
#include <math.h>

typedef __attribute__((ext_vector_type(2))) float v2f;
typedef __attribute__((ext_vector_type(8))) float v8f;

// ---------- helpers ----------
__device__ __forceinline__ unsigned f2ord(float f) {
    unsigned u = __float_as_uint(f);
    return (u & 0x80000000u) ? ~u : (u | 0x80000000u);
}
__device__ __forceinline__ float ord2f(unsigned u) {
    return (u & 0x80000000u) ? __uint_as_float(u & 0x7FFFFFFFu)
                             : __uint_as_float(~u);
}

// ---------- fill ----------
__global__ void gnn_fill_u32(unsigned* p, unsigned v, size_t count) {
    size_t i = (size_t)blockIdx.x * blockDim.x + threadIdx.x;
    size_t stride = (size_t)gridDim.x * blockDim.x;
    for (; i < count; i += stride) p[i] = v;
}

// ---------- fp32 WMMA GEMM:  C[n,M] = A[n,K] @ B[K,M] (+bias)(+elu) ----------
// M in {64,128}, K % 4 == 0. blockDim = 256 (8 waves). Each wave owns a 16x16 C tile.
__global__ void gnn_gemm_wmma(const float* __restrict__ A, const float* __restrict__ Bm,
                              const float* __restrict__ bias, float* __restrict__ C,
                              int n, int K, int M, int act) {
    extern __shared__ float lds[];
    const int colTiles = M >> 4;            // 8 or 4
    const int rowTiles = 8 / colTiles;      // 1 or 2
    const int rowsPerBlock = rowTiles * 16; // 16 or 32
    const int row0 = blockIdx.x * rowsPerBlock;
    const int ldk = K + 4;                  // pad to avoid LDS bank conflicts

    // stage A tile (coalesced)
    const int total = rowsPerBlock * K;
    for (int idx = threadIdx.x; idx < total; idx += blockDim.x) {
        int r = idx / K, c = idx - r * K;
        int gr = row0 + r;
        lds[r * ldk + c] = (gr < n) ? A[(size_t)gr * K + c] : 0.f;
    }
    __syncthreads();

    const int w    = threadIdx.x >> 5;
    const int lane = threadIdx.x & 31;
    const int laneM = lane & 15;
    const int kh    = lane >> 4;     // 0 for lanes 0-15, 1 for 16-31
    const int tileR = w / colTiles;
    const int tileC = w - tileR * colTiles;
    const int col   = tileC * 16 + laneM;

    const int aBase = (tileR * 16 + laneM) * ldk;
    const int kh2 = kh * 2;

    v8f acc = {};
    for (int k0 = 0; k0 < K; k0 += 4) {
        int ka = k0 + kh2;           // VGPR v holds K = 2*kh + v
        v2f av, bv;
        av.x = lds[aBase + ka];
        av.y = lds[aBase + ka + 1];
        const float* bp = Bm + (size_t)ka * M + col;
        bv.x = bp[0];
        bv.y = bp[M];
        acc = __builtin_amdgcn_wmma_f32_16x16x4_f32(
            false, av, false, bv, (short)0, acc, false, false);
    }

    float bvv = bias ? bias[col] : 0.f;
#pragma unroll
    for (int r = 0; r < 8; ++r) {
        int row = row0 + tileR * 16 + r + 8 * kh; // lanes 16-31 hold M = 8+r
        if (row < n) {
            float v = acc[r] + bvv;
            if (act) v = (v > 0.f) ? v : (expf(v) - 1.f); // ELU
            C[(size_t)row * M + col] = v;
        }
    }
}

// ---------- per-node attention logits: as/ad [n,H] ----------
// wave per node; supports H<=2 with C=64
__global__ void gnn_alpha_prep(const float* __restrict__ h,
                               const float* __restrict__ asrc,
                               const float* __restrict__ adst,
                               float* __restrict__ outs, float* __restrict__ outd,
                               int n, int HC, int Cdim, int H) {
    int wid  = (blockIdx.x * blockDim.x + threadIdx.x) >> 5;
    int lane = threadIdx.x & 31;
    if (wid >= n) return;
    const float* hp = h + (size_t)wid * HC;
    float sa0 = 0.f, sa1 = 0.f, da0 = 0.f, da1 = 0.f;
    for (int j = lane; j < HC; j += 32) {
        float hv = hp[j];
        float as = asrc[j], ad = adst[j];
        if (j < Cdim) { sa0 += hv * as; da0 += hv * ad; }
        else          { sa1 += hv * as; da1 += hv * ad; }
    }
#pragma unroll
    for (int off = 16; off; off >>= 1) {
        sa0 += __shfl_xor(sa0, off);
        da0 += __shfl_xor(da0, off);
        sa1 += __shfl_xor(sa1, off);
        da1 += __shfl_xor(da1, off);
    }
    if (lane == 0) {
        outs[wid * H + 0] = sa0;
        outd[wid * H + 0] = da0;
        if (H > 1) { outs[wid * H + 1] = sa1; outd[wid * H + 1] = da1; }
    }
}

// ---------- edge softmax pass 1: segment max (ordered-uint atomicMax) ----------
__global__ void gnn_edge_max(const long long* __restrict__ ei, int E, int EE,
                             const float* __restrict__ as_, const float* __restrict__ ad_,
                             unsigned* __restrict__ m, int H) {
    int i = blockIdx.x * blockDim.x + threadIdx.x;
    if (i >= EE) return;
    int s, d;
    if (i < E) { s = (int)ei[i]; d = (int)ei[E + i]; }
    else       { s = i - E; d = s; }
    for (int h = 0; h < H; ++h) {
        float e = as_[s * H + h] + ad_[d * H + h];
        e = (e > 0.f) ? e : 0.2f * e;  // leaky relu
        atomicMax(&m[d * H + h], f2ord(e));
    }
}

// ---------- edge softmax pass 2: exp + segment denom, stash p ----------
__global__ void gnn_edge_exp(const long long* __restrict__ ei, int E, int EE,
                             const float* __restrict__ as_, const float* __restrict__ ad_,
                             const unsigned* __restrict__ m, float* __restrict__ denom,
                             float* __restrict__ p, int H) {
    int i = blockIdx.x * blockDim.x + threadIdx.x;
    if (i >= EE) return;
    int s, d;
    if (i < E) { s = (int)ei[i]; d = (int)ei[E + i]; }
    else       { s = i - E; d = s; }
    for (int h = 0; h < H; ++h) {
        float e = as_[s * H + h] + ad_[d * H + h];
        e = (e > 0.f) ? e : 0.2f * e;
        float pe = expf(e - ord2f(m[d * H + h]));
        p[(size_t)i * H + h] = pe;
        atomicAdd(&denom[d * H + h], pe);
    }
}

// ---------- edge aggregate: out[dst] += alpha * h[src]  (wave per edge) ----------
__global__ void gnn_edge_aggregate(const long long* __restrict__ ei, int E, int EE,
                                   const float* __restrict__ hsrc,
                                   const float* __restrict__ p,
                                   const float* __restrict__ denom,
                                   float* __restrict__ out,
                                   int HC, int Cdim, int H) {
    int wid  = (int)(((size_t)blockIdx.x * blockDim.x + threadIdx.x) >> 5);
    int lane = threadIdx.x & 31;
    if (wid >= EE) return;
    int s, d;
    if (wid < E) { s = (int)ei[wid]; d = (int)ei[E + wid]; }
    else         { s = wid - E; d = s; }
    int per = HC >> 5;            // 4 (layer1) or 2 (layer2)
    int c0  = lane * per;
    int head = c0 / Cdim;         // uniform over this lane's elements
    float w = p[(size_t)wid * H + head] / (denom[d * H + head] + 1e-16f);
    const float* hs = hsrc + (size_t)s * HC + c0;
    float* op = out + (size_t)d * HC + c0;
    for (int j = 0; j < per; ++j) atomicAdd(op + j, w * hs[j]);
}

// ---------- elementwise bias (+optional ELU) ----------
__global__ void gnn_bias_act(const float* __restrict__ in, const float* __restrict__ bias,
                             float* __restrict__ out, size_t total, int M, int act) {
    size_t i = (size_t)blockIdx.x * blockDim.x + threadIdx.x;
    if (i >= total) return;
    float v = in[i] + bias[i % M];
    if (act) v = (v > 0.f) ? v : (expf(v) - 1.f);
    out[i] = v;
}

// ---------- column sum of t1 [n,64] into acc[64] ----------
__global__ void gnn_col_sum(const float* __restrict__ t1, float* __restrict__ acc, int n) {
    __shared__ float sh[256];
    int col = threadIdx.x & 63;
    int sub = threadIdx.x >> 6;                    // 0..3
    int rowsPer = blockDim.x >> 6;                 // 4
    float sum = 0.f;
    for (int r = blockIdx.x * rowsPer + sub; r < n; r += gridDim.x * rowsPer)
        sum += t1[(size_t)r * 64 + col];
    sh[threadIdx.x] = sum;
    __syncthreads();
    if (threadIdx.x < 64) {
        float s = sh[threadIdx.x] + sh[threadIdx.x + 64] +
                  sh[threadIdx.x + 128] + sh[threadIdx.x + 192];
        atomicAdd(&acc[threadIdx.x], s);
    }
}

// ---------- final head: out = (acc/n) @ W_lin2 + b_lin2 ----------
__global__ void gnn_final_head(const float* __restrict__ acc, const float* __restrict__ W2,
                               const float* __restrict__ b2, float* __restrict__ out, int n) {
    int j = threadIdx.x;
    if (j >= 64) return;
    float inv = 1.f / (float)n;
    float s = b2[j];
    for (int k = 0; k < 64; ++k) s += (acc[k] * inv) * W2[k * 64 + j];
    out[j] = s;
}

extern "C" void kernel_launch(void* const* d_in, const int* in_sizes, int n_in,
                              void* d_out, int out_size, void* d_ws, size_t ws_size,
                              hipStream_t stream) {
    const float*     x      = (const float*)d_in[0];
    const long long* ei     = (const long long*)d_in[1];
    const float*     W1     = (const float*)d_in[2];
    const float*     a_src1 = (const float*)d_in[3];
    const float*     a_dst1 = (const float*)d_in[4];
    const float*     b1     = (const float*)d_in[5];
    const float*     W2     = (const float*)d_in[6];
    const float*     a_src2 = (const float*)d_in[7];
    const float*     a_dst2 = (const float*)d_in[8];
    const float*     b2     = (const float*)d_in[9];
    const float*     W_lin1 = (const float*)d_in[10];
    const float*     b_lin1 = (const float*)d_in[11];
    const float*     W_lin2 = (const float*)d_in[12];
    const float*     b_lin2 = (const float*)d_in[13];
    float*           out    = (float*)d_out;

    const int N  = in_sizes[0] / 256;   // 100000
    const int E  = in_sizes[1] / 2;     // 1600000
    const int EE = E + N;               // with self loops

    // workspace layout (floats)
    float* A   = (float*)d_ws;                    // N*128  (h1, later hfeat, later t1)
    float* Bf  = A  + (size_t)N * 128;            // N*128  (out1, later h2|out2)
    float* P   = Bf + (size_t)N * 128;            // EE*2   (edge exp values)
    float* as_ = P  + (size_t)EE * 2;             // N*2
    float* ad_ = as_ + (size_t)N * 2;             // N*2
    float* m_  = ad_ + (size_t)N * 2;             // N*2 (ordered-uint max)
    float* dn_ = m_  + (size_t)N * 2;             // N*2 (denoms)
    float* acc = dn_ + (size_t)N * 2;             // 64

    // ordered-uint encoding of -inf: ~(0xFF800000) = 0x007FFFFF
    const unsigned NEG_INF_ORD = 0x007FFFFFu;
    const int TB = 256;
    dim3 blk(TB);

    // ---- Layer 1: h1 = x @ W1  (N x 256 x 128) ----
    {
        int grid = (N + 15) / 16;
        size_t ldsB = 16 * (256 + 4) * sizeof(float);
        gnn_gemm_wmma<<<grid, blk, ldsB, stream>>>(x, W1, nullptr, A, N, 256, 128, 0);
    }
    gnn_alpha_prep<<<(N + 7) / 8, blk, 0, stream>>>(A, a_src1, a_dst1, as_, ad_, N, 128, 64, 2);

    gnn_fill_u32<<<1024, blk, 0, stream>>>((unsigned*)m_, NEG_INF_ORD, (size_t)N * 2);
    gnn_fill_u32<<<1024, blk, 0, stream>>>((unsigned*)dn_, 0u, (size_t)N * 2);
    gnn_fill_u32<<<4096, blk, 0, stream>>>((unsigned*)Bf, 0u, (size_t)N * 128);

    {
        int grid = (EE + TB - 1) / TB;
        gnn_edge_max<<<grid, blk, 0, stream>>>(ei, E, EE, as_, ad_, (unsigned*)m_, 2);
        gnn_edge_exp<<<grid, blk, 0, stream>>>(ei, E, EE, as_, ad_, (unsigned*)m_, dn_, P, 2);
        int gridW = (EE + 7) / 8;  // wave per edge, 8 waves/block
        gnn_edge_aggregate<<<gridW, blk, 0, stream>>>(ei, E, EE, A, P, dn_, Bf, 128, 64, 2);
    }
    // hfeat = elu(out1 + b1) -> A
    {
        size_t total = (size_t)N * 128;
        gnn_bias_act<<<(int)((total + TB - 1) / TB), blk, 0, stream>>>(Bf, b1, A, total, 128, 1);
    }

    // ---- Layer 2: h2 = hfeat @ W2  (N x 128 x 64) ----
    float* h2   = Bf;                    // N*64
    float* out2 = Bf + (size_t)N * 64;   // N*64
    {
        int grid = (N + 31) / 32;
        size_t ldsB = 32 * (128 + 4) * sizeof(float);
        gnn_gemm_wmma<<<grid, blk, ldsB, stream>>>(A, W2, nullptr, h2, N, 128, 64, 0);
    }
    gnn_alpha_prep<<<(N + 7) / 8, blk, 0, stream>>>(h2, a_src2, a_dst2, as_, ad_, N, 64, 64, 1);

    gnn_fill_u32<<<1024, blk, 0, stream>>>((unsigned*)m_, NEG_INF_ORD, (size_t)N);
    gnn_fill_u32<<<1024, blk, 0, stream>>>((unsigned*)dn_, 0u, (size_t)N);
    gnn_fill_u32<<<4096, blk, 0, stream>>>((unsigned*)out2, 0u, (size_t)N * 64);

    {
        int grid = (EE + TB - 1) / TB;
        gnn_edge_max<<<grid, blk, 0, stream>>>(ei, E, EE, as_, ad_, (unsigned*)m_, 1);
        gnn_edge_exp<<<grid, blk, 0, stream>>>(ei, E, EE, as_, ad_, (unsigned*)m_, dn_, P, 1);
        int gridW = (EE + 7) / 8;
        gnn_edge_aggregate<<<gridW, blk, 0, stream>>>(ei, E, EE, h2, P, dn_, out2, 64, 64, 1);
    }
    // hfin = out2 + b2 (no act) -> h2 region
    {
        size_t total = (size_t)N * 64;
        gnn_bias_act<<<(int)((total + TB - 1) / TB), blk, 0, stream>>>(out2, b2, h2, total, 64, 0);
    }

    // ---- MLP: t1 = elu(hfin @ W_lin1 + b_lin1) -> A ----
    {
        int grid = (N + 31) / 32;
        size_t ldsB = 32 * (64 + 4) * sizeof(float);
        gnn_gemm_wmma<<<grid, blk, ldsB, stream>>>(h2, W_lin1, b_lin1, A, N, 64, 64, 1);
    }

    // ---- mean pool fused with last linear: out = (sum(t1)/N) @ W_lin2 + b_lin2 ----
    gnn_fill_u32<<<1, 64, 0, stream>>>((unsigned*)acc, 0u, 64);
    gnn_col_sum<<<512, blk, 0, stream>>>(A, acc, N);
    gnn_final_head<<<1, 64, 0, stream>>>(acc, W_lin2, b_lin2, out, N);
}